// Sparse_TensorProduct_9715216023820
// MI455X (gfx1250) — compile-verified
//
#include <hip/hip_runtime.h>
#include <hip/hip_bf16.h>
#include <math.h>

typedef float v2f __attribute__((ext_vector_type(2)));
typedef float v8f __attribute__((ext_vector_type(8)));

// ---------------- constants describing the tensor product ----------------
// irreps: 128x0e+128x1o+128x2e ; x1/out row = 1152 f32 ; x2 row = 9 f32
// K-dimension (padded to 36): concat of per-path l1 components:
// p: (i1,i2,io):  0:(0,0,0) 1:(0,1,1) 2:(0,2,2) 3:(1,0,1) 4:(1,1,0) 5:(1,1,2)
//                 6:(1,2,1) 7:(2,0,2) 8:(2,1,1) 9:(2,2,0) 10:(2,2,2)
__constant__ int c_kP[36]    = {0,1,2, 3,3,3, 4,4,4, 5,5,5, 6,6,6,
                                7,7,7,7,7, 8,8,8,8,8, 9,9,9,9,9, 10,10,10,10,10, 0};
__constant__ int c_kBase[36] = {0,0,0, 128,129,130, 128,129,130, 128,129,130, 128,129,130,
                                512,513,514,515,516, 512,513,514,515,516,
                                512,513,514,515,516, 512,513,514,515,516, 0};
__constant__ int c_kD[36]    = {1,1,1, 3,3,3, 3,3,3, 3,3,3, 3,3,3,
                                5,5,5,5,5, 5,5,5,5,5, 5,5,5,5,5, 5,5,5,5,5, 1};

// ---------------- CDNA5 async VMEM -> LDS helpers ----------------
__device__ __forceinline__ void async_b128(void* lds, const void* g) {
    asm volatile("global_load_async_to_lds_b128 %0, %1, off"
                 :: "v"((uint32_t)(uintptr_t)lds), "v"((uint64_t)(uintptr_t)g)
                 : "memory");
}
__device__ __forceinline__ void async_b32(void* lds, const void* g) {
    asm volatile("global_load_async_to_lds_b32 %0, %1, off"
                 :: "v"((uint32_t)(uintptr_t)lds), "v"((uint64_t)(uintptr_t)g)
                 : "memory");
}
__device__ __forceinline__ void wait_async0() {
#if __has_builtin(__builtin_amdgcn_s_wait_asynccnt)
    __builtin_amdgcn_s_wait_asynccnt(0);
#else
    asm volatile("s_wait_asynccnt 0x0" ::: "memory");
#endif
}

// ---------------- init kernel: build C3[K=36][n=9][j=9] in d_ws ----------------
// M2[K][n] = sum_j C3[K][n][j] * x2[j]  (includes alpha path normalization)
__global__ void init_c3_kernel(float* __restrict__ C3) {
    if (blockIdx.x != 0 || threadIdx.x != 0) return;
    const float S = 0.70710678118654752f;   // 1/sqrt(2)
    const float Tq = 0.40824829046386302f;  // 1/sqrt(6)
    const float R3 = 0.57735026918962576f;  // 1/sqrt(3)
    const float R5 = 0.44721359549995794f;  // 1/sqrt(5)
    // orthonormal symmetric-traceless basis (e3nn l=2 real order: xy,yz,z2,xz,x2-y2)
    float B[5][3][3] = {
        {{0,S,0},{S,0,0},{0,0,0}},
        {{0,0,0},{0,0,S},{0,S,0}},
        {{-Tq,0,0},{0,-Tq,0},{0,0,2*Tq}},
        {{0,0,S},{0,0,0},{S,0,0}},
        {{S,0,0},{0,-S,0},{0,0,0}},
    };
    const int ax[3] = {1, 2, 0}; // e3nn l=1 real order (m=-1,0,1) -> (y,z,x)
    for (int i = 0; i < 36 * 9 * 9; ++i) C3[i] = 0.f;
    // 2-2-2 coupling: T[a][b][c] = tr(Ba Bb Bc), Frobenius-normalized
    float T[5][5][5];
    float nrm2 = 0.f;
    for (int a = 0; a < 5; ++a)
        for (int b = 0; b < 5; ++b)
            for (int c = 0; c < 5; ++c) {
                float acc = 0.f;
                for (int p = 0; p < 3; ++p)
                    for (int q = 0; q < 3; ++q)
                        for (int r = 0; r < 3; ++r)
                            acc += B[a][p][q] * B[b][q][r] * B[c][r][p];
                T[a][b][c] = acc;
                nrm2 += acc * acc;
            }
    const float inv222 = 1.f / sqrtf(nrm2);

    const int d1[11]    = {1,1,1,3,3,3,3,5,5,5,5};
    const int d2[11]    = {1,3,5,1,3,3,5,1,3,5,5};
    const int dO[11]    = {1,3,5,3,1,5,3,5,3,1,5};
    const int Koff[11]  = {0,1,2,3,6,9,12,15,20,25,30};
    const int off2p[11] = {0,1,4,0,1,1,4,0,1,4,4};   // x2 block offset of i2
    const int offop[11] = {0,1,4,1,0,4,1,4,1,0,4};   // out block offset of io
    const float a0 = 0.57735026918962576f;           // sqrt(1/3)
    const float a1 = 0.86602540378443865f;           // sqrt(3/4)
    const float a2 = 1.11803398874989485f;           // sqrt(5/4)
    const float alphap[11] = {a0,a1,a2,a1,a0,a2,a1,a2,a1,a0,a2};

    for (int p = 0; p < 11; ++p)
        for (int i = 0; i < d1[p]; ++i)
            for (int j = 0; j < d2[p]; ++j)
                for (int k = 0; k < dO[p]; ++k) {
                    float v = 0.f;
                    switch (p) {
                        case 0:  v = 1.f; break;
                        case 1:  v = (j == k) ? R3 : 0.f; break;
                        case 2:  v = (j == k) ? R5 : 0.f; break;
                        case 3:  v = (i == k) ? R3 : 0.f; break;
                        case 4:  v = (i == j) ? R3 : 0.f; break;
                        case 5:  v = B[k][ax[i]][ax[j]] * R5; break;
                        case 6:  v = B[j][ax[i]][ax[k]] * R5; break;
                        case 7:  v = (i == k) ? R5 : 0.f; break;
                        case 8:  v = B[i][ax[j]][ax[k]] * R5; break;
                        case 9:  v = (i == j) ? R5 : 0.f; break;
                        case 10: v = T[i][j][k] * inv222; break;
                    }
                    C3[((Koff[p] + i) * 9 + (offop[p] + k)) * 9 + (off2p[p] + j)] =
                        alphap[p] * v;
                }
}

// ---------------- main kernel ----------------
// per sample: out(128x9) = (Xrep .* W)(128x36) @ M2(36x16)  via V_WMMA_F32_16X16X4_F32
// Double-buffered: async-prefetch sample z+stride into LDS while computing sample z.
__global__ __launch_bounds__(256)
void tp_uvu_kernel(const float* __restrict__ x1g, const float* __restrict__ x2g,
                   const float* __restrict__ wg, float* __restrict__ outg,
                   const float* __restrict__ C3g, int Z) {
    __shared__ float x1s[2][1152];
    __shared__ float x2s[2][16];
    __shared__ float outs[1152];
    __shared__ float M2s[36 * 16];
    __shared__ float c3s[36 * 9 * 9];

    const int tid  = threadIdx.x;
    const int lane = tid & 31;
    const int wid  = tid >> 5;        // u-tile 0..7 (fixed per wave)
    const int n    = lane & 15;       // A row (u_local), B/D column
    const int hi   = lane >> 4;
    const int u    = wid * 16 + n;    // fixed channel row for A fragments

    // per-lane A-fragment metadata (sample-invariant): 18 in-row indices + weights
    int   aAddr[18];
    float aW[18];
#pragma unroll
    for (int t = 0; t < 18; ++t) {
        int K = 4 * (t >> 1) + 2 * hi + (t & 1);
        if (K < 35) {
            aAddr[t] = c_kBase[K] + u * c_kD[K];
            aW[t]    = wg[c_kP[K] * 128 + u];
        } else {
            aAddr[t] = 0;
            aW[t]    = 0.f;   // K=35 zero pad
        }
    }

    // preload constant C3 into LDS; zero padded columns of M2 once
    for (int idx = tid; idx < 36 * 9 * 9; idx += 256) c3s[idx] = C3g[idx];
    for (int idx = tid; idx < 36 * 16; idx += 256)    M2s[idx] = 0.f;

    // prologue: async-prefetch first sample into buffer 0
    const int z0 = blockIdx.x;
    if (z0 < Z) {
        for (int idx = tid; idx < 288; idx += 256)
            async_b128(&x1s[0][idx * 4], x1g + (size_t)z0 * 1152 + idx * 4);
        if (tid < 9)
            async_b32(&x2s[0][tid], x2g + (size_t)z0 * 9 + tid);
    }
    __syncthreads();   // c3s / M2s padding ready

    int cur = 0;
    for (int z = z0; z < Z; z += gridDim.x, cur ^= 1) {
        // wait for this wave's async fills, then make all waves' fills visible
        wait_async0();
        __syncthreads();

        // prefetch next sample into the other buffer (overlaps with compute)
        const int zn = z + (int)gridDim.x;
        const int nxt = cur ^ 1;
        if (zn < Z) {
            for (int idx = tid; idx < 288; idx += 256)
                async_b128(&x1s[nxt][idx * 4], x1g + (size_t)zn * 1152 + idx * 4);
            if (tid < 9)
                async_b32(&x2s[nxt][tid], x2g + (size_t)zn * 9 + tid);
        }

        // M2[K][n] = C3[K][n][:] . x2  (324 dots of length 9 across the block)
        for (int e = tid; e < 36 * 9; e += 256) {
            int K  = e / 9;
            int nn = e - K * 9;
            const float* crow = &c3s[e * 9];
            float acc = 0.f;
#pragma unroll
            for (int j = 0; j < 9; ++j) acc += crow[j] * x2s[cur][j];
            M2s[K * 16 + nn] = acc;
        }
        __syncthreads();

        // 9 chained f32 WMMAs: C += A(16x4) x B(4x16)
        const float* x1c = x1s[cur];
        v8f cacc = {};
#pragma unroll
        for (int s = 0; s < 9; ++s) {
            v2f a, b;
            a.x = aW[2 * s]     * x1c[aAddr[2 * s]];
            a.y = aW[2 * s + 1] * x1c[aAddr[2 * s + 1]];
            int row = 4 * s + 2 * hi;
            b.x = M2s[row * 16 + n];
            b.y = M2s[row * 16 + 16 + n];
            cacc = __builtin_amdgcn_wmma_f32_16x16x4_f32(
                false, a, false, b, (short)0, cacc, false, false);
        }

        // scatter D into native output layout in LDS
        if (n < 9) {
            int base, d, kl;
            if (n == 0)     { base = 0;   d = 1; kl = 0;     }
            else if (n < 4) { base = 128; d = 3; kl = n - 1; }
            else            { base = 512; d = 5; kl = n - 4; }
            int u0 = wid * 16 + 8 * hi;
#pragma unroll
            for (int r = 0; r < 8; ++r)
                outs[base + (u0 + r) * d + kl] = cacc[r];
        }
        __syncthreads();

        // coalesced b128 store of out row (fire-and-forget; next-iter barrier
        // plus the compiler's dscnt drain cover the outs WAR hazard)
        float4*       dst   = (float4*)(outg + (size_t)z * 1152);
        const float4* outs4 = (const float4*)outs;
        for (int idx = tid; idx < 288; idx += 256) dst[idx] = outs4[idx];
    }
}

extern "C" void kernel_launch(void* const* d_in, const int* in_sizes, int n_in,
                              void* d_out, int out_size, void* d_ws, size_t ws_size,
                              hipStream_t stream) {
    const float* x1 = (const float*)d_in[0];
    const float* x2 = (const float*)d_in[1];
    const float* w  = (const float*)d_in[2];
    float* out      = (float*)d_out;
    float* C3       = (float*)d_ws;   // 36*9*9 floats = 11664 B

    int Z = in_sizes[0] / 1152;

    init_c3_kernel<<<1, 1, 0, stream>>>(C3);

    int nblk = (Z + 15) / 16;
    if (nblk > 2048) nblk = 2048;
    if (nblk < 1) nblk = 1;
    tp_uvu_kernel<<<nblk, 256, 0, stream>>>(x1, x2, w, out, C3, Z);
}